// PatchesLayerIter_67473936220598
// MI455X (gfx1250) — compile-verified
//
#include <hip/hip_runtime.h>
#include <hip/hip_bf16.h>
#include <stdint.h>

// Patch extraction [64,384,384,3] f32 -> [64,576,16,16,3] f32.
// Pure permutation, HBM-bound (~226 MB total => ~9.7us floor @ 23.3 TB/s).
// CDNA5 path: async global->LDS scatter (permutation folded into LDS address),
// then async LDS->global fully-contiguous store. Tracked via ASYNCcnt.

#define IMG      384
#define PBLK     24            // 384/16 patch blocks per axis
#define JH       2             // split each 16-row stripe into 2 j-halves
#define TILE_F4  2304          // 16 rows * 12 jblk * 12 float4  (36 KB)
#define WG       256
#define PER_THR  (TILE_F4 / WG)  // 9

__global__ __launch_bounds__(WG)
void patches_async_kernel(const float* __restrict__ in, float* __restrict__ out) {
    // Tile staged in LDS in OUTPUT order: [j_local(12)][row(16)][c4(12)] of float4.
    __shared__ float4 tile[TILE_F4];   // 36864 bytes

    const unsigned w     = blockIdx.x;            // 0 .. 64*24*2-1
    const unsigned b     = w / (PBLK * JH);
    const unsigned rem   = w % (PBLK * JH);
    const unsigned iblk  = rem / JH;
    const unsigned jhalf = rem % JH;
    const unsigned t     = threadIdx.x;

    // Input byte base of this stripe: row (b, iblk*16), starting at pixel jhalf*192.
    // One image row = 384 px * 12 B = 4608 B; j-half offset = 192 px * 12 B = 2304 B.
    const unsigned long long in_base =
        (unsigned long long)(uintptr_t)in
        + (unsigned long long)((b * (unsigned)IMG + iblk * 16u) * (unsigned)IMG) * 12ull
        + (unsigned long long)jhalf * 2304ull;

    // Output: tiles are exactly contiguous; tile w starts at byte w*36864.
    const unsigned long long out_base =
        (unsigned long long)(uintptr_t)out + (unsigned long long)w * 36864ull;

    // ---- Phase 1: async load, global-contiguous reads, LDS scatter into
    //      output order. Each row r contributes a contiguous 2304-B run. ----
    #pragma unroll
    for (int m = 0; m < PER_THR; ++m) {
        unsigned li  = t + (unsigned)m * WG;      // 0 .. 2303
        unsigned r   = li / 144u;                 // image row within stripe
        unsigned k   = li - r * 144u;             // float4 within the 2304-B run
        unsigned jl  = k / 12u;                   // patch column (local)
        unsigned c4  = k - jl * 12u;              // float4 within patch row
        unsigned voff = r * 4608u + k * 16u;      // global byte offset (16B aligned)
        unsigned lds  = (unsigned)(uintptr_t)&tile[(jl * 16u + r) * 12u + c4];
        asm volatile("global_load_async_to_lds_b128 %0, %1, %2"
                     :: "v"(lds), "v"(voff), "s"(in_base) : "memory");
    }

    // All async loads of this wave landed in LDS, then sync across waves
    // (a wave stores LDS regions written by other waves).
    asm volatile("s_wait_asynccnt 0x0" ::: "memory");
    __syncthreads();

    // ---- Phase 2: async store, LDS is already in output order; global
    //      destination is one fully-contiguous 36 KB block. ----
    #pragma unroll
    for (int m = 0; m < PER_THR; ++m) {
        unsigned si   = t + (unsigned)m * WG;
        unsigned voff = si * 16u;
        unsigned lds  = (unsigned)(uintptr_t)&tile[si];
        asm volatile("global_store_async_from_lds_b128 %0, %1, %2"
                     :: "v"(voff), "v"(lds), "s"(out_base) : "memory");
    }

    // Drain outstanding async stores (S_ENDPGM also waits, but be explicit).
    asm volatile("s_wait_asynccnt 0x0" ::: "memory");
}

extern "C" void kernel_launch(void* const* d_in, const int* in_sizes, int n_in,
                              void* d_out, int out_size, void* d_ws, size_t ws_size,
                              hipStream_t stream) {
    (void)in_sizes; (void)n_in; (void)d_ws; (void)ws_size; (void)out_size;
    const float* pixels = (const float*)d_in[0];
    float* out = (float*)d_out;
    const int grid = 64 * PBLK * JH;   // 3072 workgroups, 8 waves each
    patches_async_kernel<<<grid, WG, 0, stream>>>(pixels, out);
}